// SpGATClassification_79706003079272
// MI455X (gfx1250) — compile-verified
//
#include <hip/hip_runtime.h>
#include <hip/hip_bf16.h>
#include <math.h>

#define NFEAT 512
#define NHID 64
#define NHEADS 8
#define NCLASS 40
#define LRELU_ALPHA 0.2f

typedef __attribute__((ext_vector_type(2))) float v2f;
typedef __attribute__((ext_vector_type(8))) float v8f;

__device__ __forceinline__ void atomicAddF32(float* p, float v) {
  // lowers to global_atomic_add_f32 (no return -> STOREcnt)
  __hip_atomic_fetch_add(p, v, __ATOMIC_RELAXED, __HIP_MEMORY_SCOPE_AGENT);
}

// ---------------------------------------------------------------------------
// zero fill (accumulator region must start at 0 before the atomic scatters)
// ---------------------------------------------------------------------------
__global__ __launch_bounds__(256) void gat_zero_f32(float* __restrict__ p, size_t count) {
  size_t i = (size_t)blockIdx.x * blockDim.x + threadIdx.x;
  size_t stride = (size_t)gridDim.x * blockDim.x;
  for (; i < count; i += stride) p[i] = 0.0f;
}

// ---------------------------------------------------------------------------
// Layer-1 GEMM: h1[n, head*64+f] = x[n,:] @ W[head][:, f]   (exact fp32 WMMA)
// block = 128 threads (4 waves); each wave: 16 rows x 64 cols (4 C-tiles).
// Out-of-range rows are clamped to row n-1 (safe loads, duplicate compute);
// all stores are guarded by rr < n, so no zero-masking of A is needed and
// the inner loop is branch-free.
// Fused epilogue: alphaS[n,head] = h1_row . a[head][:64],
//                 alphaD[n,head] = h1_row . a[head][64:]
// ---------------------------------------------------------------------------
__global__ __launch_bounds__(128) void gat_gemm1_wmma(
    const float* __restrict__ x, const float* __restrict__ W,
    const float* __restrict__ a, float* __restrict__ h1,
    float* __restrict__ alphaS, float* __restrict__ alphaD, int n)
{
  const int head = blockIdx.y;
  const int wave = threadIdx.x >> 5;
  const int lane = threadIdx.x & 31;
  const int half = lane >> 4;
  const int l16  = lane & 15;

  const int rowA = blockIdx.x * 64 + wave * 16 + l16;     // A-matrix row (M = l16)
  const int rowClamped = rowA < n ? rowA : (n - 1);
  const float* xrow = x + (size_t)rowClamped * NFEAT;
  const float* Wh   = W + (size_t)head * (NFEAT * NHID);

  v8f c0 = {}, c1 = {}, c2 = {}, c3 = {};

  #pragma unroll 4
  for (int k = 0; k < NFEAT; k += 4) {
    const int kk = k + 2 * half;                          // A layout: half selects K pair
    const v2f av = *(const v2f*)(xrow + kk);              // unconditional b64 load
    __builtin_prefetch(xrow + k + 32, 0, 3);

    const float* w0 = Wh + (size_t)kk * NHID;             // B row kk
    const float* w1 = w0 + NHID;                          // B row kk+1
    v2f b0, b1, b2, b3;
    b0.x = w0[l16];      b0.y = w1[l16];
    b1.x = w0[l16 + 16]; b1.y = w1[l16 + 16];
    b2.x = w0[l16 + 32]; b2.y = w1[l16 + 32];
    b3.x = w0[l16 + 48]; b3.y = w1[l16 + 48];

    c0 = __builtin_amdgcn_wmma_f32_16x16x4_f32(false, av, false, b0, (short)0, c0, false, false);
    c1 = __builtin_amdgcn_wmma_f32_16x16x4_f32(false, av, false, b1, (short)0, c1, false, false);
    c2 = __builtin_amdgcn_wmma_f32_16x16x4_f32(false, av, false, b2, (short)0, c2, false, false);
    c3 = __builtin_amdgcn_wmma_f32_16x16x4_f32(false, av, false, b3, (short)0, c3, false, false);
  }

  // C/D layout: vgpr r -> row rBase+r, lane half adds 8; col = tile*16 + l16
  const int rBase = blockIdx.x * 64 + wave * 16 + 8 * half;
  const float* ah = a + head * 2 * NHID;
  const float s0 = ah[l16],       s1 = ah[l16 + 16];
  const float s2 = ah[l16 + 32],  s3 = ah[l16 + 48];
  const float d0 = ah[NHID + l16],      d1 = ah[NHID + l16 + 16];
  const float d2 = ah[NHID + l16 + 32], d3 = ah[NHID + l16 + 48];

  #pragma unroll
  for (int r = 0; r < 8; ++r) {
    const int rr = rBase + r;
    if (rr < n) {
      float* hrow = h1 + (size_t)rr * (NHEADS * NHID) + head * NHID;
      hrow[l16]      = c0[r];
      hrow[l16 + 16] = c1[r];
      hrow[l16 + 32] = c2[r];
      hrow[l16 + 48] = c3[r];
    }
    float ss = c0[r] * s0 + c1[r] * s1 + c2[r] * s2 + c3[r] * s3;
    float sd = c0[r] * d0 + c1[r] * d1 + c2[r] * d2 + c3[r] * d3;
    #pragma unroll
    for (int m = 8; m >= 1; m >>= 1) {
      ss += __shfl_xor(ss, m, 16);
      sd += __shfl_xor(sd, m, 16);
    }
    if (l16 == 0 && rr < n) {
      alphaS[(size_t)rr * NHEADS + head] = ss;
      alphaD[(size_t)rr * NHEADS + head] = sd;
    }
  }
}

// ---------------------------------------------------------------------------
// Layer-1 edge scatter: one wave32 per edge.
//   w[h] = exp(-leakyrelu(alphaS[src,h] + alphaD[dst,h]))
//   den1[src,h]       += w[h]
//   num1[src, h*64+f] += w[h] * h1[dst, h*64+f]
// lane covers f = 32*i + lane, i = 0..15; head(i) = i>>1 (wave-uniform per i)
// ---------------------------------------------------------------------------
__global__ __launch_bounds__(256) void gat_edge1(
    const int* __restrict__ ei, int etot,
    const float* __restrict__ h1,
    const float* __restrict__ alphaS, const float* __restrict__ alphaD,
    float* __restrict__ num1, float* __restrict__ den1)
{
  const int e = blockIdx.x * 8 + (threadIdx.x >> 5);
  if (e >= etot) return;
  const int lane = threadIdx.x & 31;
  const int s = ei[e];
  const int d = ei[etot + e];

  float wloc = 0.0f;
  if (lane < NHEADS) {
    float lg = alphaS[(size_t)s * NHEADS + lane] + alphaD[(size_t)d * NHEADS + lane];
    float lr = lg > 0.0f ? lg : LRELU_ALPHA * lg;
    wloc = __expf(-lr);
    atomicAddF32(&den1[(size_t)s * NHEADS + lane], wloc);
  }

  const float* hd = h1 + (size_t)d * (NHEADS * NHID);
  float* ns = num1 + (size_t)s * (NHEADS * NHID);
  #pragma unroll
  for (int i = 0; i < 16; ++i) {
    const float w = __shfl(wloc, i >> 1, 32);
    const int f = i * 32 + lane;
    atomicAddF32(&ns[f], w * hd[f]);
  }
}

// ---------------------------------------------------------------------------
// Layer-1 finalize (in place): h2 = elu(num1 / den1)
// ---------------------------------------------------------------------------
__global__ __launch_bounds__(256) void gat_finalize1(
    float* __restrict__ num1, const float* __restrict__ den1, int n)
{
  const size_t total = (size_t)n * (NHEADS * NHID);
  size_t idx = (size_t)blockIdx.x * blockDim.x + threadIdx.x;
  size_t stride = (size_t)gridDim.x * blockDim.x;
  for (; idx < total; idx += stride) {
    const size_t node = idx >> 9;           // / 512
    const int f = (int)(idx & 511);
    const int head = f >> 6;
    float v = num1[idx] / den1[node * NHEADS + head];
    num1[idx] = v > 0.0f ? v : (__expf(v) - 1.0f);
  }
}

// ---------------------------------------------------------------------------
// Layer-2 GEMM: hout[n, c] = h2[n,:] @ W_out[:, c], c padded to 48 (3 tiles).
// Fused alpha epilogue with a_out (masked to c < 40). hout row stride = 64.
// ---------------------------------------------------------------------------
__global__ __launch_bounds__(128) void gat_gemm2_wmma(
    const float* __restrict__ h2, const float* __restrict__ Wout,
    const float* __restrict__ aout, float* __restrict__ hout,
    float* __restrict__ alphaS, float* __restrict__ alphaD, int n)
{
  const int wave = threadIdx.x >> 5;
  const int lane = threadIdx.x & 31;
  const int half = lane >> 4;
  const int l16  = lane & 15;

  const int rowA = blockIdx.x * 64 + wave * 16 + l16;
  const int rowClamped = rowA < n ? rowA : (n - 1);
  const float* xrow = h2 + (size_t)rowClamped * (NHEADS * NHID);

  v8f c0 = {}, c1 = {}, c2 = {};
  const int col2 = l16 + 32;
  const bool c2ok = col2 < NCLASS;

  #pragma unroll 4
  for (int k = 0; k < NHEADS * NHID; k += 4) {
    const int kk = k + 2 * half;
    const v2f av = *(const v2f*)(xrow + kk);              // unconditional b64 load
    __builtin_prefetch(xrow + k + 32, 0, 3);

    const float* w0 = Wout + (size_t)kk * NCLASS;
    const float* w1 = w0 + NCLASS;
    v2f b0, b1, b2;
    b0.x = w0[l16];      b0.y = w1[l16];
    b1.x = w0[l16 + 16]; b1.y = w1[l16 + 16];
    b2.x = c2ok ? w0[col2] : 0.0f;
    b2.y = c2ok ? w1[col2] : 0.0f;

    c0 = __builtin_amdgcn_wmma_f32_16x16x4_f32(false, av, false, b0, (short)0, c0, false, false);
    c1 = __builtin_amdgcn_wmma_f32_16x16x4_f32(false, av, false, b1, (short)0, c1, false, false);
    c2 = __builtin_amdgcn_wmma_f32_16x16x4_f32(false, av, false, b2, (short)0, c2, false, false);
  }

  const int rBase = blockIdx.x * 64 + wave * 16 + 8 * half;
  const float s0 = aout[l16];
  const float s1 = aout[l16 + 16];
  const float s2 = c2ok ? aout[col2] : 0.0f;
  const float d0 = aout[NCLASS + l16];
  const float d1 = aout[NCLASS + l16 + 16];
  const float d2 = c2ok ? aout[NCLASS + col2] : 0.0f;

  #pragma unroll
  for (int r = 0; r < 8; ++r) {
    const int rr = rBase + r;
    if (rr < n) {
      float* hrow = hout + (size_t)rr * 64;
      hrow[l16]      = c0[r];
      hrow[l16 + 16] = c1[r];
      hrow[col2]     = c2[r];                 // cols 40..47 are zeros, never read
    }
    float ss = c0[r] * s0 + c1[r] * s1 + c2[r] * s2;
    float sd = c0[r] * d0 + c1[r] * d1 + c2[r] * d2;
    #pragma unroll
    for (int m = 8; m >= 1; m >>= 1) {
      ss += __shfl_xor(ss, m, 16);
      sd += __shfl_xor(sd, m, 16);
    }
    if (l16 == 0 && rr < n) {
      alphaS[rr] = ss;
      alphaD[rr] = sd;
    }
  }
}

// ---------------------------------------------------------------------------
// Layer-2 edge scatter: one wave32 per edge, 40 features (stride-64 rows).
// ---------------------------------------------------------------------------
__global__ __launch_bounds__(256) void gat_edge2(
    const int* __restrict__ ei, int etot,
    const float* __restrict__ hout,
    const float* __restrict__ alphaS, const float* __restrict__ alphaD,
    float* __restrict__ num2, float* __restrict__ den2)
{
  const int e = blockIdx.x * 8 + (threadIdx.x >> 5);
  if (e >= etot) return;
  const int lane = threadIdx.x & 31;
  const int s = ei[e];
  const int d = ei[etot + e];

  float w = 0.0f;
  if (lane == 0) {
    float lg = alphaS[s] + alphaD[d];
    float lr = lg > 0.0f ? lg : LRELU_ALPHA * lg;
    w = __expf(-lr);
    atomicAddF32(&den2[s], w);
  }
  w = __shfl(w, 0, 32);

  const float* hd = hout + (size_t)d * 64;
  float* ns = num2 + (size_t)s * 64;
  atomicAddF32(&ns[lane], w * hd[lane]);
  if (lane < NCLASS - 32) {
    atomicAddF32(&ns[32 + lane], w * hd[32 + lane]);
  }
}

// ---------------------------------------------------------------------------
// Layer-2 finalize: out = log_softmax(elu(num2/den2)), one wave32 per node.
// ---------------------------------------------------------------------------
__global__ __launch_bounds__(256) void gat_finalize2(
    const float* __restrict__ num2, const float* __restrict__ den2,
    float* __restrict__ out, int n)
{
  const int node = blockIdx.x * 8 + (threadIdx.x >> 5);
  if (node >= n) return;
  const int lane = threadIdx.x & 31;
  const float den = den2[node];
  const float* nrow = num2 + (size_t)node * 64;

  float v0 = nrow[lane] / den;
  v0 = v0 > 0.0f ? v0 : (__expf(v0) - 1.0f);
  const bool hi = lane < (NCLASS - 32);
  float v1 = -1e30f;
  if (hi) {
    v1 = nrow[32 + lane] / den;
    v1 = v1 > 0.0f ? v1 : (__expf(v1) - 1.0f);
  }

  float m = fmaxf(v0, v1);
  #pragma unroll
  for (int sh = 16; sh >= 1; sh >>= 1) m = fmaxf(m, __shfl_xor(m, sh, 32));

  float ssum = __expf(v0 - m) + (hi ? __expf(v1 - m) : 0.0f);
  #pragma unroll
  for (int sh = 16; sh >= 1; sh >>= 1) ssum += __shfl_xor(ssum, sh, 32);

  const float lse = m + __logf(ssum);
  float* orow = out + (size_t)node * NCLASS;
  orow[lane] = v0 - lse;
  if (hi) orow[32 + lane] = v1 - lse;
}

// ---------------------------------------------------------------------------
// Host-side orchestration
// ---------------------------------------------------------------------------
extern "C" void kernel_launch(void* const* d_in, const int* in_sizes, int n_in,
                              void* d_out, int out_size, void* d_ws, size_t ws_size,
                              hipStream_t stream) {
  const float* x     = (const float*)d_in[0];   // [N, 512]
  const int*   ei    = (const int*)  d_in[1];   // [2, E+N]
  const float* W     = (const float*)d_in[2];   // [8, 512, 64]
  const float* a     = (const float*)d_in[3];   // [8, 128]
  const float* W_out = (const float*)d_in[4];   // [512, 40]
  const float* a_out = (const float*)d_in[5];   // [80]
  float* out = (float*)d_out;                   // [N, 40]

  const int n    = in_sizes[0] / NFEAT;
  const int etot = in_sizes[1] / 2;
  const size_t nn = (size_t)n;

  // workspace layout (floats); accumulators [num1 den1 num2 den2] contiguous
  float* ws   = (float*)d_ws;
  float* h1   = ws;                    // 512*nn
  float* aS1  = h1   + 512 * nn;       // 8*nn
  float* aD1  = aS1  + 8 * nn;         // 8*nn
  float* hout = aD1  + 8 * nn;         // 64*nn (stride 64, 40 used)
  float* aS2  = hout + 64 * nn;        // nn
  float* aD2  = aS2  + nn;             // nn
  float* num1 = aD2  + nn;             // 512*nn (zero region begins here)
  float* den1 = num1 + 512 * nn;       // 8*nn
  float* num2 = den1 + 8 * nn;         // 64*nn
  float* den2 = num2 + 64 * nn;        // nn (zero region ends here)
  const size_t zcount = (512 + 8 + 64 + 1) * nn;

  // 1) zero the accumulators
  {
    int blocks = (int)((zcount + 255) / 256);
    if (blocks > 65535) blocks = 65535;   // grid-stride
    gat_zero_f32<<<blocks, 256, 0, stream>>>(num1, zcount);
  }
  // 2) layer-1 GEMM (WMMA f32) + fused alpha
  {
    dim3 grid((n + 63) / 64, NHEADS);
    gat_gemm1_wmma<<<grid, 128, 0, stream>>>(x, W, a, h1, aS1, aD1, n);
  }
  // 3) layer-1 edge scatter
  gat_edge1<<<(etot + 7) / 8, 256, 0, stream>>>(ei, etot, h1, aS1, aD1, num1, den1);
  // 4) layer-1 finalize (elu(num/den) in place -> layer-2 input)
  {
    size_t total = nn * (NHEADS * NHID);
    int blocks = (int)((total + 255) / 256);
    if (blocks > 65535) blocks = 65535;
    gat_finalize1<<<blocks, 256, 0, stream>>>(num1, den1, n);
  }
  // 5) layer-2 GEMM (WMMA f32) + fused alpha
  gat_gemm2_wmma<<<(n + 63) / 64, 128, 0, stream>>>(num1, W_out, a_out, hout, aS2, aD2, n);
  // 6) layer-2 edge scatter
  gat_edge2<<<(etot + 7) / 8, 256, 0, stream>>>(ei, etot, hout, aS2, aD2, num2, den2);
  // 7) layer-2 finalize + log-softmax
  gat_finalize2<<<(n + 7) / 8, 256, 0, stream>>>(num2, den2, out, n);
}